// DetectionPostprocess_49881750176163
// MI455X (gfx1250) — compile-verified
//
#include <hip/hip_runtime.h>
#include <math.h>

// Problem constants from the reference (fixed by setup_inputs / CROP_SIZE).
#define DHW      24
#define NDHW     13824          // 24*24*24
#define TPB      256            // 8 wave32 waves per workgroup
#define PER_T    (NDHW / TPB)   // 54 strided elements per thread
#define NWAVE    (TPB / 32)
#define TOPK_N   60
#define NMS_N    20
#define NEGV     (-1.0e9f)
#define STRIDE_F 4.0f           // 96 / 24

// gfx1250 async global->LDS path (guarded; falls back to plain loads).
#if defined(__gfx1250__) && __has_builtin(__builtin_amdgcn_global_load_async_to_lds_b128)
#define HAVE_ASYNC_LDS 1
#else
#define HAVE_ASYNC_LDS 0
#endif

// The builtin signature (from hipcc's diagnostic) takes pointers to
// int __attribute__((vector_size(16))) in addrspace(1) / addrspace(3).
typedef int v4i __attribute__((vector_size(16)));
typedef __attribute__((address_space(1))) v4i gv4i;
typedef __attribute__((address_space(3))) v4i lv4i;

// Clang rejects direct casts that change address space in HIP, so round-trip
// through integers (CK-style). Generic LDS pointers hold the LDS byte offset
// in their low 32 bits (ISA 10.2), so truncation == addrspacecast to AS3.
__device__ __forceinline__ gv4i* to_global_v4i(const void* p) {
  return (gv4i*)(unsigned long long)p;
}
__device__ __forceinline__ lv4i* to_lds_v4i(void* p) {
  return (lv4i*)(unsigned int)(unsigned long long)p;
}

// Order-preserving float -> u32 key (monotone for all finite floats).
__device__ __forceinline__ unsigned fkey(float f) {
  unsigned u = __float_as_uint(f);
  return (u & 0x80000000u) ? ~u : (u | 0x80000000u);
}

__device__ __forceinline__ unsigned long long shflxor64(unsigned long long v, int m) {
  unsigned lo = (unsigned)__shfl_xor((int)(unsigned)v, m, 32);
  unsigned hi = (unsigned)__shfl_xor((int)(unsigned)(v >> 32), m, 32);
  return ((unsigned long long)hi << 32) | lo;
}

__global__ __launch_bounds__(TPB) void det_post_kernel(
    const float* __restrict__ cls, const float* __restrict__ shp,
    const float* __restrict__ off, float* __restrict__ out) {
  // 54 KB key tile + small scratch; fits well under the 320 KB WGP LDS pool.
  __shared__ unsigned keys[NDHW];
  __shared__ unsigned long long red[NWAVE];
  __shared__ unsigned long long gsel;
  __shared__ int topk_idx[TOPK_N];
  __shared__ float boxes[TOPK_N][6];
  __shared__ float work[TOPK_N];
  __shared__ int sel_i;
  __shared__ float sel_s;

  const int b = blockIdx.x;
  const int tid = threadIdx.x;
  const float* clsb = cls + (size_t)b * NDHW;
  float* outb = out + (size_t)b * (TOPK_N * 8);

  // dets initialized to -1 (rows >= NMS_N stay -1).
  for (int j = tid; j < TOPK_N * 8; j += TPB) outb[j] = -1.0f;

  // ---- Stage logits into LDS -------------------------------------------
#if HAVE_ASYNC_LDS
  for (int i = tid; i < NDHW / 4; i += TPB) {
    __builtin_amdgcn_global_load_async_to_lds_b128(
        to_global_v4i(clsb + 4 * i), to_lds_v4i(&keys[4 * i]), 0, 0);
  }
#if __has_builtin(__builtin_amdgcn_s_wait_asynccnt)
  __builtin_amdgcn_s_wait_asynccnt(0);
#else
  asm volatile("s_wait_asynccnt 0" ::: "memory");
#endif
#else
  for (int i = tid; i < NDHW; i += TPB) keys[i] = __float_as_uint(clsb[i]);
#endif
  __syncthreads();

  // ---- Convert to ordered keys; cache per-thread local max --------------
  // Thread t owns elements i == t (mod TPB); nobody else touches them.
  unsigned lbest = 0u;
  int lidx = tid;
#pragma unroll 6
  for (int j = 0; j < PER_T; ++j) {
    int i = tid + j * TPB;
    unsigned k = fkey(__uint_as_float(keys[i]));
    keys[i] = k;
    if (k > lbest) { lbest = k; lidx = i; }
  }

  // ---- Top-60 selection (sigmoid is monotone: select on raw logits) ----
  for (int k = 0; k < TOPK_N; ++k) {
    unsigned long long kv = ((unsigned long long)lbest << 32) | (unsigned)lidx;
#pragma unroll
    for (int m = 16; m > 0; m >>= 1) {
      unsigned long long o = shflxor64(kv, m);
      if (o > kv) kv = o;
    }
    if ((tid & 31) == 0) red[tid >> 5] = kv;
    __syncthreads();
    if (tid == 0) {
      unsigned long long m = red[0];
#pragma unroll
      for (int w = 1; w < NWAVE; ++w)
        if (red[w] > m) m = red[w];
      gsel = m;
      topk_idx[k] = (int)(unsigned)m;
    }
    __syncthreads();
    const int g = (int)(unsigned)gsel;
    if ((g & (TPB - 1)) == tid) {       // only the owner rescans its 54 keys
      keys[g] = 0u;
      lbest = 0u;
      lidx = tid;
#pragma unroll 6
      for (int j = 0; j < PER_T; ++j) {
        int i = tid + j * TPB;
        unsigned kk = keys[i];
        if (kk > lbest) { lbest = kk; lidx = i; }
      }
    }
    __syncthreads();
  }

  // ---- Build boxes only for the 60 winners (sparse gather) --------------
  if (tid < TOPK_N) {
    const int idx = topk_idx[tid];
    const float logit = clsb[idx];
    const float s = 1.0f / (1.0f + expf(-logit));
    const int z = idx / (DHW * DHW);
    const int y = (idx / DHW) % DHW;
    const int x = idx % DHW;
    const size_t base = (size_t)b * 3 * NDHW + (size_t)idx;
    boxes[tid][0] = ((float)z + off[base]) * STRIDE_F;
    boxes[tid][1] = ((float)y + off[base + NDHW]) * STRIDE_F;
    boxes[tid][2] = ((float)x + off[base + 2 * NDHW]) * STRIDE_F;
    boxes[tid][3] = shp[base];
    boxes[tid][4] = shp[base + NDHW];
    boxes[tid][5] = shp[base + 2 * NDHW];
    work[tid] = (s > 0.15f) ? s : NEGV;
  }
  __syncthreads();

  // ---- Greedy NMS, 20 rounds over 60 candidates -------------------------
  for (int k = 0; k < NMS_N; ++k) {
    if (tid == 0) {  // serial argmax over 60 values: first-max tie-break
      float m = work[0];
      int mi = 0;
      for (int t = 1; t < TOPK_N; ++t) {
        const float v = work[t];
        if (v > m) { m = v; mi = t; }
      }
      sel_i = mi;
      sel_s = m;
    }
    __syncthreads();
    const int i = sel_i;
    const float s = sel_s;
    const bool valid = s > (NEGV * 0.5f);
    if (tid < 8) {
      float v = -1.0f;
      if (valid) {
        if (tid == 0)      v = 1.0f;
        else if (tid == 1) v = s;
        else               v = boxes[i][tid - 2];
      }
      outb[k * 8 + tid] = v;
    }
    if (valid && tid < TOPK_N) {   // per-candidate 3D IoU suppression lane
      const float c0 = boxes[i][0], c1 = boxes[i][1], c2 = boxes[i][2];
      const float d0 = boxes[i][3], d1 = boxes[i][4], d2 = boxes[i][5];
      const float e0 = boxes[tid][0], e1 = boxes[tid][1], e2 = boxes[tid][2];
      const float f0 = boxes[tid][3], f1 = boxes[tid][4], f2 = boxes[tid][5];
      const float i0 = fminf(c0 + 0.5f * d0, e0 + 0.5f * f0) - fmaxf(c0 - 0.5f * d0, e0 - 0.5f * f0);
      const float i1 = fminf(c1 + 0.5f * d1, e1 + 0.5f * f1) - fmaxf(c1 - 0.5f * d1, e1 - 0.5f * f1);
      const float i2 = fminf(c2 + 0.5f * d2, e2 + 0.5f * f2) - fmaxf(c2 - 0.5f * d2, e2 - 0.5f * f2);
      const float inter = fmaxf(i0, 0.0f) * fmaxf(i1, 0.0f) * fmaxf(i2, 0.0f);
      const float v1 = d0 * d1 * d2;
      const float v2 = f0 * f1 * f2;
      const float iou = inter / (v1 + v2 - inter + 1e-8f);
      if (iou > 0.05f) work[tid] = NEGV;
    }
    __syncthreads();
    if (tid == 0) work[i] = NEGV;
    __syncthreads();
  }
}

extern "C" void kernel_launch(void* const* d_in, const int* in_sizes, int n_in,
                              void* d_out, int out_size, void* d_ws, size_t ws_size,
                              hipStream_t stream) {
  (void)n_in; (void)out_size; (void)d_ws; (void)ws_size;
  const float* cls = (const float*)d_in[0];   // (B,1,24,24,24) f32
  const float* shp = (const float*)d_in[1];   // (B,3,24,24,24) f32
  const float* off = (const float*)d_in[2];   // (B,3,24,24,24) f32
  float* out = (float*)d_out;                 // (B,60,8) f32
  const int B = in_sizes[0] / NDHW;
  det_post_kernel<<<B, TPB, 0, stream>>>(cls, shp, off, out);
}